// StockSolver_63634235457867
// MI455X (gfx1250) — compile-verified
//
#include <hip/hip_runtime.h>
#include <math.h>

typedef __attribute__((ext_vector_type(2))) float v2f;
typedef __attribute__((ext_vector_type(8))) float v8f;
typedef unsigned v4u __attribute__((ext_vector_type(4)));
typedef unsigned v8u __attribute__((ext_vector_type(8)));

constexpr int kN   = 256;   // variables per QP
constexpr int kNB  = 16;    // panel / tile width
constexpr int kNT  = kN / kNB;
constexpr int kLDA = 257;   // padded LDS row stride (odd -> conflict-free column walks)
constexpr int kThreads = 256;
constexpr int kWaves   = kThreads / 32;

// dynamic LDS: A[256*257] + 8 vectors of 256 + 8 scalars
constexpr int kSmemFloats = kN * kLDA + 8 * kN + 8;

__global__ __launch_bounds__(kThreads)
void qp_activeset_kernel(const float* __restrict__ Q,
                         const float* __restrict__ m,
                         const float* __restrict__ p,
                         const unsigned char* __restrict__ act,       // bool mask (B,N)
                         const float* __restrict__ av,                // active_values
                         const unsigned char* __restrict__ act_budget,// bool (B,)
                         const float* __restrict__ budget_w,          // (B,)
                         float* __restrict__ out)
{
    extern __shared__ float sm[];
    float* Amat = sm;                  // kN * kLDA  (holds raw Q, then masked H=2Q, then L)
    float* r0   = sm + kN * kLDA;      // rhs col 0 -> x_eq
    float* r1   = r0 + kN;             // rhs col 1 -> y
    float* sAct = r1 + kN;
    float* sAv  = sAct + kN;
    float* sM   = sAv + kN;
    float* sP   = sM + kN;
    float* red0 = sP + kN;
    float* red1 = red0 + kN;
    float* sScl = red1 + kN;           // [0]=alpha [1]=budget_active [2]=budget_w

    const int tid  = threadIdx.x;
    const int lane = tid & 31;
    const int wave = tid >> 5;
    const int hf   = lane >> 4;        // half-wave select: 0 = lanes 0-15, 1 = 16-31
    const int l16  = lane & 15;
    const int b    = blockIdx.x;
    const size_t qb = (size_t)b * kN * kN;

    // ---------------- phase 1: small per-batch vectors ----------------
    sAct[tid] = (float)act[b * kN + tid];
    sAv[tid]  = av[b * kN + tid];
    sM[tid]   = m[b * kN + tid];
    sP[tid]   = p[b * kN + tid];
    if (tid == 0) {
        sScl[1] = (float)act_budget[b];
        sScl[2] = budget_w[b];
    }

    // ---------------- phase 2: stream raw Q -> LDS via Tensor Data Mover ----------
    // One TENSOR_LOAD_TO_LDS per wave: 32x256 f32 slice, row-major, with the D#
    // pad feature inserting 1 DWORD after every 256 DWORDs -> LDS stride kLDA=257.
    // Tracked by TENSORcnt; EXEC is ignored for tensor ops (one issue per wave).
    {
        const unsigned ldsbase = __builtin_amdgcn_groupstaticsize();  // 0: no static LDS
        const unsigned w = __builtin_amdgcn_readfirstlane((unsigned)wave);
        const unsigned rows = kN / kWaves;                            // 32 rows per wave
        unsigned long long ga = (unsigned long long)(Q + qb)
                              + (unsigned long long)w * rows * kN * 4ull;
        unsigned lds_addr = ldsbase + 4u * (unsigned)(w * rows * kLDA);

        v4u g0;
        g0[0] = 1u;                                           // count = 1 valid descriptor
        g0[1] = lds_addr;                                     // LDS byte address
        g0[2] = (unsigned)ga;                                 // global addr [31:0]
        g0[3] = (unsigned)((ga >> 32) & 0x01FFFFFFull)        // global addr [56:32]
              | (2u << 30);                                   // type = 2 ("image")

        v8u g1;
        g1[0] = (2u << 16)        // data_size = 2 -> 4-byte elements
              | (1u << 20)       // pad_enable
              | (7u << 22)       // pad_interval = 7 -> every 256 DWORDs
              | (0u << 25);      // pad_amount = 0 -> 1 DWORD  => LDS stride 257
        g1[1] = (unsigned)kN << 16;      // tensor_dim0 = 256 (bits [79:48], low half)
        g1[2] = rows << 16;              // tensor_dim0 hi = 0 | tensor_dim1 = 32 (low half)
        g1[3] = (unsigned)kN << 16;      // tensor_dim1 hi = 0 | tile_dim0 = 256
        g1[4] = rows;                    // tile_dim1 = 32, tile_dim2 = 0
        g1[5] = (unsigned)kN;            // tensor_dim0_stride = 256 (bits [207:160], low)
        g1[6] = 0u;                      // stride hi = 0, tensor_dim1_stride lo = 0
        g1[7] = 0u;                      // tensor_dim1_stride hi = 0

        asm volatile("tensor_load_to_lds %0, %1" :: "s"(g0), "s"(g1) : "memory");
        __builtin_amdgcn_s_wait_tensorcnt(0);
    }
    __syncthreads();

    // ---------------- phase 3: Hav = (2Q) @ av, build both RHS ----------------
    for (int r = wave; r < kN; r += kWaves) {
        float s = 0.f;
        #pragma unroll
        for (int k = 0; k < kN / 32; ++k) {
            int c = lane + 32 * k;
            s += Amat[r * kLDA + c] * sAv[c];
        }
        #pragma unroll
        for (int off = 16; off > 0; off >>= 1) s += __shfl_xor(s, off);
        if (lane == 0) {
            float hav = 2.f * s;                       // fold H = 2Q here
            if (sAct[r] != 0.f) { r0[r] = sAv[r];           r1[r] = 0.f;   }
            else                { r0[r] = -(sM[r] + hav);   r1[r] = sP[r]; }
        }
    }
    __syncthreads();

    // ---------------- phase 4: mask + scale in place: A = masked(2Q) ----------------
    for (int idx = tid; idx < kN * kN; idx += kThreads) {
        int i = idx >> 8, j = idx & 255;
        float a = Amat[i * kLDA + j];
        bool keep = (sAct[i] == 0.f) && (sAct[j] == 0.f);
        Amat[i * kLDA + j] = keep ? 2.f * a : ((i == j) ? 1.0f : 0.0f);
    }
    __syncthreads();

    // ---------------- phase 5: blocked Cholesky (WMMA trailing SYRK) ----------------
    for (int kb = 0; kb < kNT; ++kb) {
        const int base = kb * kNB;

        // (a) 16x16 diagonal factor: wave 0, lane i caches row i in registers,
        //     column broadcasts via shuffles. Unguarded updates/stores only touch
        //     the (unused) upper triangle -> straight-line code, no exec churn.
        if (wave == 0) {
            float row[kNB];
            #pragma unroll
            for (int t = 0; t < kNB; ++t)
                row[t] = (lane < kNB) ? Amat[(base + lane) * kLDA + base + t] : 0.f;
            #pragma unroll
            for (int j = 0; j < kNB; ++j) {
                float dj = sqrtf(__shfl(row[j], j));
                float cj = (lane == j) ? dj : row[j] / dj;  // scaled col j
                row[j] = cj;
                #pragma unroll
                for (int t = j + 1; t < kNB; ++t)
                    row[t] -= cj * __shfl(cj, t);           // rank-1 update
            }
            if (lane < kNB) {
                #pragma unroll
                for (int t = 0; t < kNB; ++t)
                    Amat[(base + lane) * kLDA + base + t] = row[t];
            }
        }
        __syncthreads();

        // (b) panel TRSM: L21 = A21 * L11^{-T}, one row per thread, row in registers
        {
            int r = base + kNB + tid;
            if (r < kN) {
                float rv[kNB];
                #pragma unroll
                for (int t = 0; t < kNB; ++t) rv[t] = Amat[r * kLDA + base + t];
                #pragma unroll
                for (int j = 0; j < kNB; ++j) {
                    float s = rv[j];
                    #pragma unroll
                    for (int t = 0; t < j; ++t)
                        s -= rv[t] * Amat[(base + j) * kLDA + base + t];
                    rv[j] = s / Amat[(base + j) * kLDA + base + j];
                }
                #pragma unroll
                for (int t = 0; t < kNB; ++t) Amat[r * kLDA + base + t] = rv[t];
            }
        }
        __syncthreads();

        // (c) trailing update: C(ib,jb) -= P(ib) * P(jb)^T, ib >= jb,
        //     4 chained V_WMMA_F32_16X16X4_F32 per 16x16 tile. Tile index is
        //     wave-uniform so EXEC is all-ones around the WMMAs.
        {
            const int T = kNT - 1 - kb;
            const int ntiles = T * (T + 1) / 2;
            for (int t = wave; t < ntiles; t += kWaves) {
                int ibr = 0;
                while ((ibr + 1) * (ibr + 2) / 2 <= t) ++ibr;
                int jbr = t - ibr * (ibr + 1) / 2;
                int ib = base + kNB + ibr * kNB;
                int jb = base + kNB + jbr * kNB;

                // C tile: VGPR r holds row (r + 8*hf), col l16
                v8f c;
                #pragma unroll
                for (int r = 0; r < 8; ++r)
                    c[r] = Amat[(ib + r + 8 * hf) * kLDA + jb + l16];

                #pragma unroll
                for (int ck = 0; ck < 4; ++ck) {
                    int k0 = base + ck * 4 + 2 * hf;
                    v2f a, bb;
                    // A = -P(ib) chunk (negate product -> subtraction; f32 has no A/B NEG)
                    a[0]  = -Amat[(ib + l16) * kLDA + k0];
                    a[1]  = -Amat[(ib + l16) * kLDA + k0 + 1];
                    // B = P(jb)^T chunk (same gather pattern on P(jb))
                    bb[0] =  Amat[(jb + l16) * kLDA + k0];
                    bb[1] =  Amat[(jb + l16) * kLDA + k0 + 1];
                    c = __builtin_amdgcn_wmma_f32_16x16x4_f32(
                            false, a, false, bb, (short)0, c, false, false);
                }
                #pragma unroll
                for (int r = 0; r < 8; ++r)
                    Amat[(ib + r + 8 * hf) * kLDA + jb + l16] = c[r];
            }
        }
        __syncthreads();
    }

    // ---------------- phase 6: forward solve L z = rhs (2 RHS), blocked ----------------
    for (int kb = 0; kb < kNT; ++kb) {
        const int base = kb * kNB;
        if (tid < 2) {                      // thread 0 -> r0, thread 1 -> r1
            float* rr = tid ? r1 : r0;
            float z[kNB];
            #pragma unroll
            for (int j = 0; j < kNB; ++j) {
                float s = rr[base + j];
                #pragma unroll
                for (int t = 0; t < j; ++t)
                    s -= Amat[(base + j) * kLDA + base + t] * z[t];
                z[j] = s / Amat[(base + j) * kLDA + base + j];
            }
            #pragma unroll
            for (int j = 0; j < kNB; ++j) rr[base + j] = z[j];
        }
        __syncthreads();
        {
            int r = base + kNB + tid;
            if (r < kN) {
                float s0 = r0[r], s1 = r1[r];
                #pragma unroll
                for (int j = 0; j < kNB; ++j) {
                    float l = Amat[r * kLDA + base + j];
                    s0 -= l * r0[base + j];
                    s1 -= l * r1[base + j];
                }
                r0[r] = s0; r1[r] = s1;
            }
        }
        __syncthreads();
    }

    // ---------------- phase 7: backward solve L^T x = z (2 RHS), blocked ----------------
    for (int kb = kNT - 1; kb >= 0; --kb) {
        const int base = kb * kNB;
        if (tid < 2) {
            float* rr = tid ? r1 : r0;
            float x[kNB];
            #pragma unroll
            for (int j = kNB - 1; j >= 0; --j) {
                float s = rr[base + j];
                #pragma unroll
                for (int t = j + 1; t < kNB; ++t)
                    s -= Amat[(base + t) * kLDA + base + j] * x[t];
                x[j] = s / Amat[(base + j) * kLDA + base + j];
            }
            #pragma unroll
            for (int j = 0; j < kNB; ++j) rr[base + j] = x[j];
        }
        __syncthreads();
        for (int r = tid; r < base; r += kThreads) {   // rows above this block
            float s0 = r0[r], s1 = r1[r];
            #pragma unroll
            for (int j = 0; j < kNB; ++j) {
                float l = Amat[(base + j) * kLDA + r];  // (L^T)[r][base+j]
                s0 -= l * r0[base + j];
                s1 -= l * r1[base + j];
            }
            r0[r] = s0; r1[r] = s1;
        }
        __syncthreads();
    }

    // ---------------- phase 8: budget-plane projection ----------------
    {
        float peff = (sAct[tid] != 0.f) ? 0.f : sP[tid];
        red0[tid] = peff * r1[tid];        // p_eff . y
        red1[tid] = sP[tid] * r0[tid];     // p . x_eq
    }
    __syncthreads();
    #pragma unroll
    for (int s = kThreads / 2; s > 0; s >>= 1) {
        if (tid < s) { red0[tid] += red0[tid + s]; red1[tid] += red1[tid + s]; }
        __syncthreads();
    }
    if (tid == 0) {
        float denom = fmaxf(red0[0], 1e-30f);
        sScl[0] = (sScl[1] != 0.f) ? (sScl[2] - red1[0]) / denom : 0.f;
    }
    __syncthreads();

    out[(size_t)b * kN + tid] = r0[tid] + sScl[0] * r1[tid];
}

extern "C" void kernel_launch(void* const* d_in, const int* in_sizes, int n_in,
                              void* d_out, int out_size, void* d_ws, size_t ws_size,
                              hipStream_t stream) {
    const float*         Q   = (const float*)d_in[0];
    const float*         m   = (const float*)d_in[1];
    const float*         p   = (const float*)d_in[2];
    const unsigned char* act = (const unsigned char*)d_in[3]; // jnp.bool_ -> 1 byte
    const float*         av  = (const float*)d_in[4];
    const unsigned char* ab  = (const unsigned char*)d_in[5]; // jnp.bool_ -> 1 byte
    const float*         bw  = (const float*)d_in[6];
    float*               out = (float*)d_out;

    const int B = in_sizes[6];                 // budget_w has B elements
    const size_t smem = (size_t)kSmemFloats * sizeof(float);   // ~271 KB < 320 KB/WGP

    hipFuncSetAttribute(reinterpret_cast<const void*>(qp_activeset_kernel),
                        hipFuncAttributeMaxDynamicSharedMemorySize, (int)smem);
    qp_activeset_kernel<<<B, kThreads, smem, stream>>>(Q, m, p, act, av, ab, bw, out);
}